// PureCRF_31533649887496
// MI455X (gfx1250) — compile-verified
//
#include <hip/hip_runtime.h>
#include <math.h>

#define NUM_TAGS 48
#define BATCH    512
#define SEQ      2048
#define TILE     16
#define NT       3      // 48 / 16 column tiles
#define KBLK     12     // 48 / 4  k-blocks for f32 WMMA (K=4)
#define EPAD     52     // padded LDS row stride (floats): 208B, 16B-aligned, bank-safe

typedef __attribute__((ext_vector_type(2))) float v2f;
typedef __attribute__((ext_vector_type(8))) float v8f;
typedef __attribute__((ext_vector_type(4))) int   v4i;

#if defined(__gfx1250__) &&                                             \
    __has_builtin(__builtin_amdgcn_global_load_async_to_lds_b128) &&    \
    __has_builtin(__builtin_amdgcn_s_wait_asynccnt)
#define USE_ASYNC_LDS 1
typedef __attribute__((address_space(1))) v4i v4i_g;   // global v4i
typedef __attribute__((address_space(3))) v4i v4i_l;   // LDS v4i
#else
#define USE_ASYNC_LDS 0
#endif

// ---------------------------------------------------------------------------
// Forward algorithm (partition function) via exp-domain WMMA matmuls.
// One wave32 handles M=16 batches; B = exp(transitions) is 48x48, kept in
// register fragments for all 2048 steps. Each step: 36x v_wmma_f32_16x16x4_f32.
// Emissions double-buffered in LDS via async global->LDS copies (ASYNCcnt).
// ---------------------------------------------------------------------------
__global__ __launch_bounds__(32) void crf_forward_wmma(
    const float* __restrict__ emissions,        // [B, S, T]
    const float* __restrict__ transitions,      // [T, T]
    const float* __restrict__ start_transitions,// [T]
    const float* __restrict__ end_transitions,  // [T]
    const float* __restrict__ mask,             // [B, S]
    float* __restrict__ partition_out)          // [B]
{
    __shared__ float Els[NUM_TAGS * NUM_TAGS];  // exp(transitions), row-major
    __shared__ float expAls[TILE][EPAD];        // A-fragment staging (16 rows x 48)
#if USE_ASYNC_LDS
    __shared__ float ebuf[2][TILE][EPAD];       // double-buffered emission slab
#endif

    const int lane = threadIdx.x;   // 0..31
    const int h    = lane >> 4;     // half-wave: rows 0-7 vs rows 8-15 in C/D layout
    const int n    = lane & 15;     // column-within-tile (C/D/B), row index (A)
    const int b0   = blockIdx.x * TILE;

    // --- exp(transitions) into LDS (once) ---
    for (int idx = lane; idx < NUM_TAGS * NUM_TAGS; idx += 32)
        Els[idx] = __expf(transitions[idx]);
    __syncthreads();

    // --- B fragments (4x16 f32): vgpr0 = K {0|2}, vgpr1 = K {1|3} per half ---
    v2f Bfrag[KBLK][NT];
    for (int kb = 0; kb < KBLK; ++kb) {
        const int k0 = 4 * kb + 2 * h;
        for (int nt = 0; nt < NT; ++nt) {
            Bfrag[kb][nt].x = Els[(k0 + 0) * NUM_TAGS + nt * 16 + n];
            Bfrag[kb][nt].y = Els[(k0 + 1) * NUM_TAGS + nt * 16 + n];
        }
    }

    // --- alpha held in C/D layout: alphaR[nt][r] = alpha[row 8h+r][col nt*16+n] ---
    float alphaR[NT][8];
    for (int nt = 0; nt < NT; ++nt) {
        const int j = nt * 16 + n;
        for (int r = 0; r < 8; ++r) {
            const int b = b0 + 8 * h + r;
            alphaR[nt][r] = start_transitions[j] +
                            emissions[((size_t)b * SEQ) * NUM_TAGS + j];
        }
    }

#if USE_ASYNC_LDS
    // Loop-invariant chunk mapping for the async emission copies:
    // 16 rows x 192B = 192 x 16B chunks, 32 lanes x 6 instructions per step.
    int crow[6], coff[6];           // chunk -> (batch row, 16B offset within row)
    const float* gsrc[6];           // global source for step t (bumped each step)
    for (int i = 0; i < 6; ++i) {
        const int c = i * 32 + lane;
        crow[i] = c / 12;
        coff[i] = c % 12;
        gsrc[i] = emissions + ((size_t)(b0 + crow[i]) * SEQ + 1) * NUM_TAGS +
                  coff[i] * 4;      // primed for t = 1
    }
    // Issue the slab for t = 1 into buffer 1.
    for (int i = 0; i < 6; ++i)
        __builtin_amdgcn_global_load_async_to_lds_b128(
            (v4i_g*)gsrc[i],
            (v4i_l*)&ebuf[1][crow[i]][coff[i] * 4], 0, 0);
#endif

    for (int t = 1; t < SEQ; ++t) {
#if USE_ASYNC_LDS
        // Kick off next step's emission slab (overlaps this whole iteration).
        if (t + 1 < SEQ) {
            for (int i = 0; i < 6; ++i) {
                gsrc[i] += NUM_TAGS;   // advance one time step
                __builtin_amdgcn_global_load_async_to_lds_b128(
                    (v4i_g*)gsrc[i],
                    (v4i_l*)&ebuf[(t + 1) & 1][crow[i]][coff[i] * 4], 0, 0);
            }
        }
#else
        float emitv[NT][8];
        for (int nt = 0; nt < NT; ++nt) {
            const int j = nt * 16 + n;
            for (int r = 0; r < 8; ++r) {
                const int b = b0 + 8 * h + r;
                emitv[nt][r] = emissions[((size_t)b * SEQ + t) * NUM_TAGS + j];
            }
        }
        if (t + 1 < SEQ) {
            for (int r = 0; r < 8; ++r)
                __builtin_prefetch(
                    &emissions[((size_t)(b0 + 8 * h + r) * SEQ + t + 1) * NUM_TAGS + n], 0, 0);
        }
#endif
        float mval[8];
        for (int r = 0; r < 8; ++r)
            mval[r] = mask[(size_t)(b0 + 8 * h + r) * SEQ + t];

        // Per-row max over 48 states (3 local tiles, then across the 16-lane half).
        float mx[8];
        for (int r = 0; r < 8; ++r) {
            float m = fmaxf(fmaxf(alphaR[0][r], alphaR[1][r]), alphaR[2][r]);
            m = fmaxf(m, __shfl_xor(m, 1, 16));
            m = fmaxf(m, __shfl_xor(m, 2, 16));
            m = fmaxf(m, __shfl_xor(m, 4, 16));
            m = fmaxf(m, __shfl_xor(m, 8, 16));
            mx[r] = m;
        }

        // exp(alpha - rowmax) -> LDS (D-layout write), reread in A-layout.
        for (int nt = 0; nt < NT; ++nt)
            for (int r = 0; r < 8; ++r)
                expAls[8 * h + r][nt * 16 + n] = __expf(alphaR[nt][r] - mx[r]);
        __syncthreads();

        // A fragments (16x4 f32): lane n = row n; K = 4*kb + 2*h + {0,1}.
        v2f Afrag[KBLK];
        for (int kb = 0; kb < KBLK; ++kb) {
            Afrag[kb].x = expAls[n][4 * kb + 2 * h + 0];
            Afrag[kb].y = expAls[n][4 * kb + 2 * h + 1];
        }
        __syncthreads();

        // D = expAlpha (16x48) x expTrans (48x48): 12 K-steps x 3 N-tiles.
        v8f acc[NT] = {};
        for (int kb = 0; kb < KBLK; ++kb)
            for (int nt = 0; nt < NT; ++nt)
                acc[nt] = __builtin_amdgcn_wmma_f32_16x16x4_f32(
                    false, Afrag[kb], false, Bfrag[kb][nt],
                    (short)0, acc[nt], false, false);

#if USE_ASYNC_LDS
        // Slab for step t is the oldest 6 outstanding async copies.
        if (t + 1 < SEQ)
            __builtin_amdgcn_s_wait_asynccnt(6);
        else
            __builtin_amdgcn_s_wait_asynccnt(0);
        float emitv[NT][8];
        for (int nt = 0; nt < NT; ++nt)
            for (int r = 0; r < 8; ++r)
                emitv[nt][r] = ebuf[t & 1][8 * h + r][nt * 16 + n];
#endif

        // alpha' = rowmax + log(dot) + emit, blended by mask.
        for (int nt = 0; nt < NT; ++nt)
            for (int r = 0; r < 8; ++r) {
                const float na = mx[r] + __logf(acc[nt][r]) + emitv[nt][r];
                alphaR[nt][r] = na * mval[r] + alphaR[nt][r] * (1.0f - mval[r]);
            }
    }

    // partition[b] = logsumexp_j(alpha[b,j] + end_transitions[j])
    for (int r = 0; r < 8; ++r) {
        const float v0 = alphaR[0][r] + end_transitions[ 0 + n];
        const float v1 = alphaR[1][r] + end_transitions[16 + n];
        const float v2 = alphaR[2][r] + end_transitions[32 + n];
        float m = fmaxf(fmaxf(v0, v1), v2);
        m = fmaxf(m, __shfl_xor(m, 1, 16));
        m = fmaxf(m, __shfl_xor(m, 2, 16));
        m = fmaxf(m, __shfl_xor(m, 4, 16));
        m = fmaxf(m, __shfl_xor(m, 8, 16));
        float s = __expf(v0 - m) + __expf(v1 - m) + __expf(v2 - m);
        s += __shfl_xor(s, 1, 16);
        s += __shfl_xor(s, 2, 16);
        s += __shfl_xor(s, 4, 16);
        s += __shfl_xor(s, 8, 16);
        if (n == 0)
            partition_out[b0 + 8 * h + r] = m + __logf(s);
    }
}

// ---------------------------------------------------------------------------
// Gold-path score, one block per batch element.
// ---------------------------------------------------------------------------
__global__ __launch_bounds__(256) void crf_score(
    const float* __restrict__ emissions,
    const float* __restrict__ transitions,
    const float* __restrict__ start_transitions,
    const float* __restrict__ end_transitions,
    const int*   __restrict__ tags,
    const float* __restrict__ mask,
    float* __restrict__ score_out)
{
    __shared__ float red[256];
    __shared__ float redm[256];
    const int b = blockIdx.x, tid = threadIdx.x;
    const float* em = emissions + (size_t)b * SEQ * NUM_TAGS;
    const int*   tg = tags      + (size_t)b * SEQ;
    const float* mk = mask      + (size_t)b * SEQ;

    float partial = 0.0f, msum = 0.0f;
    for (int t = tid; t < SEQ; t += 256) {
        msum += mk[t];
        if (t >= 1) {
            const int cur = tg[t], prev = tg[t - 1];
            partial += (transitions[prev * NUM_TAGS + cur] +
                        em[(size_t)t * NUM_TAGS + cur]) * mk[t];
        } else {
            partial += start_transitions[tg[0]] + em[tg[0]];
        }
    }
    red[tid] = partial;
    redm[tid] = msum;
    __syncthreads();
    for (int s = 128; s > 0; s >>= 1) {
        if (tid < s) { red[tid] += red[tid + s]; redm[tid] += redm[tid + s]; }
        __syncthreads();
    }
    if (tid == 0) {
        const int last = (int)(redm[0] + 0.5f) - 1;
        score_out[b] = red[0] + end_transitions[tg[last]];
    }
}

// ---------------------------------------------------------------------------
// out = mean(partition - score)
// ---------------------------------------------------------------------------
__global__ __launch_bounds__(256) void crf_final(
    const float* __restrict__ partition_in,
    const float* __restrict__ score_in,
    float* __restrict__ out)
{
    __shared__ float red[256];
    const int tid = threadIdx.x;
    float s = 0.0f;
    for (int b = tid; b < BATCH; b += 256)
        s += partition_in[b] - score_in[b];
    red[tid] = s;
    __syncthreads();
    for (int k = 128; k > 0; k >>= 1) {
        if (tid < k) red[tid] += red[tid + k];
        __syncthreads();
    }
    if (tid == 0) out[0] = red[0] / (float)BATCH;
}

extern "C" void kernel_launch(void* const* d_in, const int* in_sizes, int n_in,
                              void* d_out, int out_size, void* d_ws, size_t ws_size,
                              hipStream_t stream) {
    const float* emissions         = (const float*)d_in[0];
    const float* transitions       = (const float*)d_in[1];
    const float* start_transitions = (const float*)d_in[2];
    const float* end_transitions   = (const float*)d_in[3];
    const int*   tags              = (const int*)d_in[4];
    const float* mask              = (const float*)d_in[5];

    float* partition_ws = (float*)d_ws;          // [BATCH]
    float* score_ws     = partition_ws + BATCH;  // [BATCH]
    float* out          = (float*)d_out;

    crf_forward_wmma<<<BATCH / TILE, 32, 0, stream>>>(
        emissions, transitions, start_transitions, end_transitions, mask,
        partition_ws);
    crf_score<<<BATCH, 256, 0, stream>>>(
        emissions, transitions, start_transitions, end_transitions, tags, mask,
        score_ws);
    crf_final<<<1, 256, 0, stream>>>(partition_ws, score_ws, out);
}